// Involution2D_13357348291353
// MI455X (gfx1250) — compile-verified
//
#include <hip/hip_runtime.h>
#include <hip/hip_bf16.h>

// Problem constants (from reference setup_inputs)
constexpr int Bn  = 4;            // batch
constexpr int C   = 256;          // channels
constexpr int Cr  = 64;           // reduced channels (C/4)
constexpr int Hh  = 56;
constexpr int Ww  = 56;
constexpr int HW  = Hh * Ww;      // 3136 (divisible by 64)
constexpr int Kk  = 7;            // involution kernel size
constexpr int KK  = Kk * Kk;      // 49
constexpr int G   = 16;           // groups
constexpr int KKG = KK * G;       // 784 (divisible by 16)
constexpr int P   = 3;            // padding

typedef float v2f __attribute__((ext_vector_type(2)));
typedef float v8f __attribute__((ext_vector_type(8)));

// ---------------------------------------------------------------------------
// fp32 WMMA GEMM:  Out[b] (MxN) = A (MxK, row-major weights, shared across
// batch) * X[b] (KxN, row-major) + bias, optional ReLU.
//
// One wave32 per block computes a 16x64 output strip (4 adjacent 16x16
// N-tiles) with V_WMMA_F32_16X16X4_F32, reusing each A fragment 4x.
// Grid: (N/64, M/16, B) -> no div/mod tile decomposition, all loop-invariant
// address math hoisted by the compiler.
//
// ISA fragment layouts (CDNA5 ISA):
//   A 16x4 f32 : lanes 0-15 -> M=lane,    v0:K=0 v1:K=1
//                lanes16-31 -> M=lane-16, v0:K=2 v1:K=3
//   B 4x16 f32 : lanes 0-15 -> N=lane,    v0:K=0 v1:K=1
//                lanes16-31 -> N=lane-16, v0:K=2 v1:K=3
//   C/D 16x16  : VGPR v, lanes 0-15 -> M=v,   N=lane
//                        lanes16-31 -> M=v+8, N=lane-16
// ---------------------------------------------------------------------------
template <bool RELU>
__global__ __launch_bounds__(32)
void wmma_gemm_f32(const float* __restrict__ A,
                   const float* __restrict__ bias,
                   const float* __restrict__ X,
                   float* __restrict__ Out,
                   int M, int K, int N)
{
    const int lane = threadIdx.x;           // 0..31
    const int n0   = blockIdx.x << 6;       // 64-wide N strip
    const int m0   = blockIdx.y << 4;       // 16-wide M tile
    const int b    = blockIdx.z;

    const int lm = lane & 15;               // M (for A) or N (for B/C) index
    const int lh = lane >> 4;               // half-wave selector

    // A element (m0+lm, k + 2*lh): two contiguous K values per lane -> one
    // 8-byte-aligned float2 load per K-step ((m0+lm)*K + 2*lh is even).
    const float* __restrict__ arow  = A + (size_t)(m0 + lm) * K + (lh << 1);
    // B element (k + 2*lh, n0 + 16*j + lm): coalesced dword loads per tile.
    const float* __restrict__ bbase = X + (size_t)b * K * N
                                        + (size_t)(lh << 1) * N + (n0 + lm);

    v8f acc0 = {}, acc1 = {}, acc2 = {}, acc3 = {};

    for (int k = 0; k < K; k += 4) {
        const v2f a = *(const v2f*)(arow + k);
        const float* __restrict__ bp = bbase + (size_t)k * N;

        v2f b0; b0.x = bp[ 0];      b0.y = bp[(size_t)N +  0];
        v2f b1; b1.x = bp[16];      b1.y = bp[(size_t)N + 16];
        v2f b2; b2.x = bp[32];      b2.y = bp[(size_t)N + 32];
        v2f b3; b3.x = bp[48];      b3.y = bp[(size_t)N + 48];

        acc0 = __builtin_amdgcn_wmma_f32_16x16x4_f32(
            false, a, false, b0, (short)0, acc0, false, false);
        acc1 = __builtin_amdgcn_wmma_f32_16x16x4_f32(
            false, a, false, b1, (short)0, acc1, false, false);
        acc2 = __builtin_amdgcn_wmma_f32_16x16x4_f32(
            false, a, false, b2, (short)0, acc2, false, false);
        acc3 = __builtin_amdgcn_wmma_f32_16x16x4_f32(
            false, a, false, b3, (short)0, acc3, false, false);
    }

    // Epilogue: bias (+ReLU), coalesced stores. VGPR v -> M = m0 + 8*lh + v.
    const int mrow = m0 + (lh << 3);
    float bv[8];
#pragma unroll
    for (int v = 0; v < 8; ++v) bv[v] = bias[mrow + v];

    float* __restrict__ orow = Out + (size_t)b * M * N + (size_t)mrow * N
                                   + (n0 + lm);
#pragma unroll
    for (int v = 0; v < 8; ++v) {
        float o0 = acc0[v] + bv[v];
        float o1 = acc1[v] + bv[v];
        float o2 = acc2[v] + bv[v];
        float o3 = acc3[v] + bv[v];
        if (RELU) {
            o0 = fmaxf(o0, 0.0f); o1 = fmaxf(o1, 0.0f);
            o2 = fmaxf(o2, 0.0f); o3 = fmaxf(o3, 0.0f);
        }
        float* __restrict__ p = orow + (size_t)v * N;
        p[ 0] = o0;
        p[16] = o1;
        p[32] = o2;
        p[48] = o3;
    }
}

// ---------------------------------------------------------------------------
// Involution gather: out[b,c,h,w] = sum_{i,j} spanned[b, g*49 + i*7+j, h, w]
//                                   * x[b, c, h+i-3, w+j-3]  (zero padded)
// One thread per output element; 7x7 fully unrolled; spanned reads coalesced
// across w; x neighborhood reads served by L2/WGP$.
// ---------------------------------------------------------------------------
__global__ __launch_bounds__(256)
void involution_gather(const float* __restrict__ x,
                       const float* __restrict__ spanned,
                       float* __restrict__ out)
{
    const int idx = blockIdx.x * blockDim.x + threadIdx.x;
    constexpr int total = Bn * C * HW;
    if (idx >= total) return;

    const int w = idx % Ww;
    const int h = (idx / Ww) % Hh;
    const int c = (idx / HW) % C;
    const int b = idx / (C * HW);
    const int g = c >> 4;                   // c / (C/G) = c / 16

    const float* __restrict__ sp = spanned
        + ((size_t)b * KKG + (size_t)g * KK) * HW + (size_t)h * Ww + w;
    const float* __restrict__ xb = x + ((size_t)b * C + c) * HW;

    float acc = 0.0f;
#pragma unroll
    for (int i = 0; i < Kk; ++i) {
        const int hi = h + i - P;
        const bool hok = (unsigned)hi < (unsigned)Hh;
#pragma unroll
        for (int j = 0; j < Kk; ++j) {
            const int wj = w + j - P;
            if (hok && (unsigned)wj < (unsigned)Ww) {
                acc = fmaf(sp[(size_t)(i * Kk + j) * HW],
                           xb[(size_t)hi * Ww + wj], acc);
            }
        }
    }
    out[idx] = acc;
}

// ---------------------------------------------------------------------------
// kernel_launch: GEMM1 (reduce+ReLU) -> GEMM2 (span+bias) -> gather
// Workspace: reduced (B*Cr*HW f32 = 3.2 MB) then spanned (B*KKG*HW = 39.3 MB)
// ---------------------------------------------------------------------------
extern "C" void kernel_launch(void* const* d_in, const int* in_sizes, int n_in,
                              void* d_out, int out_size, void* d_ws, size_t ws_size,
                              hipStream_t stream)
{
    const float* x  = (const float*)d_in[0];   // (B, C, H, W)
    const float* Wr = (const float*)d_in[1];   // (Cr, C)
    const float* br = (const float*)d_in[2];   // (Cr,)
    const float* Ws = (const float*)d_in[3];   // (KKG, Cr)
    const float* bs = (const float*)d_in[4];   // (KKG,)
    float* out = (float*)d_out;                // (B, C, H, W)

    float* reduced = (float*)d_ws;                      // (B, Cr, HW)
    float* spanned = reduced + (size_t)Bn * Cr * HW;    // (B, KKG, HW)

    // GEMM1: reduced = relu(Wr @ x + br)   M=64, K=256, N=3136
    {
        dim3 grid(HW / 64, Cr / 16, Bn);                // (49, 4, 4)
        wmma_gemm_f32<true><<<grid, 32, 0, stream>>>(Wr, br, x, reduced,
                                                     Cr, C, HW);
    }

    // GEMM2: spanned = Ws @ reduced + bs   M=784, K=64, N=3136
    {
        dim3 grid(HW / 64, KKG / 16, Bn);               // (49, 49, 4)
        wmma_gemm_f32<false><<<grid, 32, 0, stream>>>(Ws, bs, reduced, spanned,
                                                      KKG, Cr, HW);
    }

    // Involution gather
    {
        const int total  = Bn * C * HW;                 // 3,211,264
        const int blocks = (total + 255) / 256;
        involution_gather<<<blocks, 256, 0, stream>>>(x, spanned, out);
    }
}